// ARqGPS_46445776339369
// MI455X (gfx1250) — compile-verified
//
#include <hip/hip_runtime.h>
#include <hip/hip_bf16.h>
#include <math.h>

typedef float v2f __attribute__((ext_vector_type(2)));
typedef float v4f __attribute__((ext_vector_type(4)));
typedef float v8f __attribute__((ext_vector_type(8)));
typedef unsigned int v4u __attribute__((ext_vector_type(4)));
typedef int v8i __attribute__((ext_vector_type(8)));
typedef int v4i __attribute__((ext_vector_type(4)));

#define B_TOT 8192
#define L_TOT 1024
#define M_DIM 128
#define D_DIM 2
#define HALF_L 512
#define SPC 16                                  // steps per TDM chunk
#define CHUNK_ELEMS (SPC * M_DIM * D_DIM)       // 4096 floats
#define CHUNK_BYTES (CHUNK_ELEMS * 4)           // 16 KB
#define NCHUNK (L_TOT / SPC)                    // 64

// ---------------------------------------------------------------------------
// eps (D, M, L) -> eps_t (L, M, D)  : per-step slice [m][d] contiguous (1 MB)
// ---------------------------------------------------------------------------
__global__ void transpose_eps(const float* __restrict__ eps,
                              float* __restrict__ eps_t) {
    int idx = blockIdx.x * 256 + threadIdx.x;          // = (d*M + m)*L + i
    int i = idx & (L_TOT - 1);
    int t = idx >> 10;                                 // d*M + m
    int m = t & (M_DIM - 1);
    int d = t >> 7;
    eps_t[(i * M_DIM + m) * D_DIM + d] = eps[idx];
}

// ---------------------------------------------------------------------------
// inputs (B, L) -> in_t (L, B)   tiled 32x32 transpose via LDS
// ---------------------------------------------------------------------------
__global__ void transpose_in(const int* __restrict__ in, int* __restrict__ out) {
    __shared__ int tile[32][33];
    int i0 = blockIdx.x * 32, b0 = blockIdx.y * 32;
    for (int j = threadIdx.y; j < 32; j += 8)
        tile[j][threadIdx.x] = in[(size_t)(b0 + j) * L_TOT + i0 + threadIdx.x];
    __syncthreads();
    for (int j = threadIdx.y; j < 32; j += 8)
        out[(size_t)(i0 + j) * B_TOT + b0 + threadIdx.x] = tile[threadIdx.x][j];
}

// ---------------------------------------------------------------------------
// TDM: async 1-D copy of one 16 KB eps chunk into LDS (uniform descriptor).
// D# group0: count=1, lds_addr, global_addr[56:0], type=2.
// D# group1: data_size=4B, tensor_dim0 huge (no OOB), tile_dim0=4096, 1 row.
// This toolchain exposes the 6-arg builtin: (g0, g1, g2, g3, g4, cpol).
// ---------------------------------------------------------------------------
__device__ __forceinline__ void tdm_load_chunk(const void* gp, unsigned lds_off) {
    unsigned long long ga = (unsigned long long)gp;
    v4u g0 = { 1u,                                   // count=1 (valid), user mode
               lds_off,                              // lds_addr (bytes)
               (unsigned)(ga & 0xffffffffu),         // global_addr[31:0]
               ((unsigned)(ga >> 32) & 0x01ffffffu)  // global_addr[56:32]
                   | 0x80000000u };                  // type=2 ("image")
    v8i g1 = { (int)0x00020000,                      // data_size=2 (4 bytes)
               0,                                    // tensor_dim0 lo16 in [31:16]=0
               0x00011000,                           // tensor_dim0 hi16=0x1000, tensor_dim1=1
               0x10000000,                           // tile_dim0 = 0x1000 (4096 elems)
               0x00000001,                           // tile_dim1 = 1
               0x10000000,                           // tensor_dim0_stride (don't care, 1 row)
               0, 0 };
    v4i gz4 = { 0, 0, 0, 0 };
    v8i gz8 = { 0, 0, 0, 0, 0, 0, 0, 0 };
    __builtin_amdgcn_tensor_load_to_lds(g0, g1, gz4, gz4, gz8, 0);
}

// ---------------------------------------------------------------------------
// Main kernel: 1 wave / workgroup, 16 batch rows per wave.
//   B-frag (registers): g[ch] = (g[row][4ch+2h], g[row][4ch+2h+1]), row=lane%16
//   A-frag (from LDS):  eps rows d=0,1 for the K=4 m-chunk
//   C rows 0,1 -> dot[row, d] land on lanes 0..15 elements 0,1 directly.
// ---------------------------------------------------------------------------
__global__ __launch_bounds__(32) void arqgps_main(
        const int* __restrict__ inp, int sI, int sB,
        const float* __restrict__ eps_t, float* __restrict__ out) {
    __shared__ float lds[2 * CHUNK_ELEMS];             // 32 KB double buffer

    // Escape the LDS array address so later memory-clobbering asm statements
    // are assumed to (re)write it -> compiler cannot fold the ds_loads of the
    // TDM-filled buffer to poison ("no stores visible" reasoning).
    {
        const float* esc = (const float*)lds;
        asm volatile("" :: "v"(esc) : "memory");
    }

    const int lane = threadIdx.x;
    const int row  = lane & 15;
    const int hh   = lane >> 4;                        // half (K offset 2h)
    const int b    = blockIdx.x * 16 + row;

    v2f g[32];
#pragma unroll
    for (int c = 0; c < 32; ++c) { g[c].x = 1.0f; g[c].y = 1.0f; }

    float res = 0.0f;
    int   c0  = 0;                                     // count of value 0 in inputs[0..i-1]

    // prefetch chunks 0 and 1
    tdm_load_chunk(eps_t, 0u);
    tdm_load_chunk(eps_t + CHUNK_ELEMS, (unsigned)CHUNK_BYTES);

    for (int k = 0; k < NCHUNK; ++k) {
        if (k == NCHUNK - 1) __builtin_amdgcn_s_wait_tensorcnt(0);
        else                 __builtin_amdgcn_s_wait_tensorcnt(1);

        // Ordering point: after this asm (may-write since lds escaped above),
        // ds_loads below must re-read the TDM-written buffer.
        unsigned base = (unsigned)((k & 1) * CHUNK_ELEMS);
        asm volatile("" : "+v"(base) :: "memory");

        for (int ii = 0; ii < SPC; ++ii) {
            const int i = k * SPC + ii;
            const int s = inp[(size_t)i * sI + (size_t)b * sB];
            const bool sel = (s != 0);

            const v4f* ep = (const v4f*)(lds + base + ii * (M_DIM * D_DIM));
            v8f acc0 = {0.f, 0.f, 0.f, 0.f, 0.f, 0.f, 0.f, 0.f};
            v8f acc1 = {0.f, 0.f, 0.f, 0.f, 0.f, 0.f, 0.f, 0.f};

#pragma unroll
            for (int ch = 0; ch < 32; ++ch) {
                v4f e = ep[ch * 2 + hh];   // {e0[m'], e1[m'], e0[m'+1], e1[m'+1]}
                v2f a;                      // A-frag: eps row (lane%16==d), junk ok for d>=2
                a.x = (row == 1) ? e.y : e.x;
                a.y = (row == 1) ? e.w : e.z;
                if (ch & 1)
                    acc1 = __builtin_amdgcn_wmma_f32_16x16x4_f32(
                            false, a, false, g[ch], (short)0, acc1, false, false);
                else
                    acc0 = __builtin_amdgcn_wmma_f32_16x16x4_f32(
                            false, a, false, g[ch], (short)0, acc0, false, false);
                // g update for next step: g *= eps[sel]
                float es0 = sel ? e.y : e.x;
                float es1 = sel ? e.w : e.z;
                g[ch].x *= es0;
                g[ch].y *= es1;
            }

            // epilogue (meaningful on lanes 0..15; lanes 16..31 compute junk, never stored)
            float lp0 = acc0[0] + acc1[0];             // dot[row, d=0]
            float lp1 = acc0[1] + acc1[1];             // dot[row, d=1]
            float lps = sel ? lp1 : lp0;
            float lpo = sel ? lp0 : lp1;
            int   no  = sel ? c0 : (i - c0);           // count of the *other* value so far
            float t   = 2.0f * (lpo - lps);
            float u   = fmaxf(t, 0.0f) + log1pf(__expf(-fabsf(t)));
            u = (no >= HALF_L) ? 0.0f : u;             // other branch renormed to -inf
            res -= 0.5f * u;
            c0 += sel ? 0 : 1;
        }

        // all ds_loads of this buffer must land before TDM overwrites it
        asm volatile("s_wait_dscnt 0" ::: "memory");
        if (k + 2 < NCHUNK)
            tdm_load_chunk(eps_t + (size_t)(k + 2) * CHUNK_ELEMS,
                           (unsigned)((k & 1) * CHUNK_BYTES));
    }

    if (lane < 16) out[b] = res;
}

// ---------------------------------------------------------------------------
extern "C" void kernel_launch(void* const* d_in, const int* in_sizes, int n_in,
                              void* d_out, int out_size, void* d_ws, size_t ws_size,
                              hipStream_t stream) {
    const int*   inputs = (const int*)d_in[0];
    const float* eps    = (const float*)d_in[1];
    float*       out    = (float*)d_out;

    float* eps_t = (float*)d_ws;
    const size_t eps_bytes = (size_t)L_TOT * M_DIM * D_DIM * sizeof(float);  // 1 MB
    const size_t in_bytes  = (size_t)B_TOT * L_TOT * sizeof(int);            // 32 MB
    int* in_t = (int*)((char*)d_ws + eps_bytes);
    const bool use_tr = (ws_size >= eps_bytes + in_bytes);

    transpose_eps<<<(L_TOT * M_DIM * D_DIM) / 256, 256, 0, stream>>>(eps, eps_t);
    if (use_tr)
        transpose_in<<<dim3(L_TOT / 32, B_TOT / 32), dim3(32, 8), 0, stream>>>(inputs, in_t);

    const int* ip = use_tr ? (const int*)in_t : inputs;
    const int  sI = use_tr ? B_TOT : 1;
    const int  sB = use_tr ? 1 : L_TOT;

    arqgps_main<<<B_TOT / 16, 32, 0, stream>>>(ip, sI, sB, eps_t, out);
}